// InnerProductFeatures_72310069395518
// MI455X (gfx1250) — compile-verified
//
#include <hip/hip_runtime.h>
#include <math.h>

// v2f: A/B fragments of V_WMMA_F32_16X16X4_F32 (16x4 f32 = 64 vals / 32 lanes = 2 VGPRs)
typedef __attribute__((ext_vector_type(2))) float v2f;
typedef __attribute__((ext_vector_type(8))) float v8f;

#define WAVES_PER_BLOCK 8
#define ROWS_PER_WAVE   16
#define CDIM            64
#define DDIM            8
#define KTOT            512   // C*D contraction length

__global__ __launch_bounds__(256) void ipf_wmma_kernel(const float* __restrict__ x,
                                                       const float* __restrict__ W,
                                                       float* __restrict__ out,
                                                       int V) {
    // WfT[n][m] = Wf[m][n] = W[c=m%64][n][d=m/64]; rows n>=8 zero (pads WMMA N=16).
    __shared__ float w_lds[16][KTOT];                     // 32 KB, shared by block
    __shared__ float s_lds[WAVES_PER_BLOCK][16][16];      // 8 KB, per-wave s tiles

    const int tid = threadIdx.x;
    for (int idx = tid; idx < 16 * KTOT; idx += 256) {
        int n = idx >> 9;          // output column k (0..15)
        int m = idx & (KTOT - 1);  // contraction index (0..511)
        int c = m & 63;
        int d = m >> 6;
        w_lds[n][m] = (n < DDIM) ? W[c * (DDIM * DDIM) + n * DDIM + d] : 0.0f;
    }
    __syncthreads();

    const int wave = tid >> 5;
    const int lane = tid & 31;
    const int half = lane >> 4;   // 0: holds K+0/K+1, 1: holds K+2/K+3
    const int l16  = lane & 15;
    const int rowbase = (blockIdx.x * WAVES_PER_BLOCK + wave) * ROWS_PER_WAVE;
    if (rowbase >= V) return;

    // ---- Phase 1: s(16x8, padded to 16x16) = x_tile(16x512) @ Wf(512x16) via WMMA f32
    const float* xrow = x + (size_t)(rowbase + l16) * KTOT;  // A: row M = lane%16
    const float* wrow = &w_lds[l16][0];                      // B: col N = lane%16
    const int off = half * 2;

    v8f acc = {};
#pragma unroll 4
    for (int kk = 0; kk < KTOT / 4; ++kk) {
        v2f a = *(const v2f*)(xrow + kk * 4 + off);   // x[row, 4kk+off .. +1]
        v2f b = *(const v2f*)(wrow + kk * 4 + off);   // WfT[col, 4kk+off .. +1]
        acc = __builtin_amdgcn_wmma_f32_16x16x4_f32(false, a, false, b,
                                                    (short)0, acc, false, false);
    }

    // Spill C tile to LDS: VGPR g holds row M = g + 8*half, column N = lane%16.
    {
        const int M0 = half * 8;
#pragma unroll
        for (int g = 0; g < 8; ++g) s_lds[wave][M0 + g][l16] = acc[g];
    }
    // LDS ops from the same wave complete in order (DScnt in-order) — safe to read back.

    // ---- Phase 2: xOx[r,i] = sum_k s[r,k] * x[r, k*64+i], then tanh.
    // Lane handles columns i = 2*lane, 2*lane+1 (32 lanes * 2 = 64 cols, 256B store/row).
#pragma unroll 2
    for (int r = 0; r < ROWS_PER_WAVE; ++r) {
        const float* xr2 = x + (size_t)(rowbase + r) * KTOT + 2 * lane;
        v2f a0 = {0.0f, 0.0f};
#pragma unroll
        for (int k = 0; k < DDIM; ++k) {
            float sk = s_lds[wave][r][k];          // LDS broadcast
            v2f  xv = *(const v2f*)(xr2 + k * CDIM);
            a0 += sk * xv;                          // rows just streamed -> L0 hit
        }
        float* op = out + (size_t)(rowbase + r) * CDIM + 2 * lane;
        op[0] = tanhf(a0.x);
        op[1] = tanhf(a0.y);
    }
}

extern "C" void kernel_launch(void* const* d_in, const int* in_sizes, int n_in,
                              void* d_out, int out_size, void* d_ws, size_t ws_size,
                              hipStream_t stream) {
    const float* x = (const float*)d_in[0];   // (V, 512) fp32
    const float* W = (const float*)d_in[1];   // (64, 8, 8) fp32
    float* out = (float*)d_out;               // (V, 64) fp32

    const int V = in_sizes[0] / KTOT;
    const int rows_per_block = WAVES_PER_BLOCK * ROWS_PER_WAVE;  // 128
    const int grid = (V + rows_per_block - 1) / rows_per_block;  // 1024 for V=131072
    ipf_wmma_kernel<<<grid, 256, 0, stream>>>(x, W, out, V);
}